// ParametricMultichannelWienerFilter_3977139716226
// MI455X (gfx1250) — compile-verified
//
#include <hip/hip_runtime.h>

typedef __attribute__((ext_vector_type(2))) float v2f;
typedef __attribute__((ext_vector_type(8))) float v8f;

#define BB 4
#define CC 8
#define FF 512
#define TT 1000

#define BETA_     1.0f
#define DIAG_REG_ 1e-6f
#define EPS_      1e-8f
#define PSD_EPS_  1e-10f

#define NBF (BB * FF)                    // 2048 (b,f) pairs
#define PSD_RAW_FLOATS (NBF * 2 * 256)   // raw 16x16 accumulators, 4 MB
#define G_OFF PSD_RAW_FLOATS             // filter matrices, 1 MB

// ---------------------------------------------------------------------------
// Stage 1: masked covariance accumulation via V_WMMA_F32_16X16X4_F32.
// One wave per (b,f). A tile rows 0..7 = re channels, 8..15 = im channels.
// B tile = A elementwise * mask (per-k broadcast). D blocks give:
//   psd_re = D[0:8,0:8] + D[8:16,8:16],  psd_im = D[8:16,0:8] - D[0:8,8:16]
// (recombination deferred to stage 2; raw accumulators stored to ws).
// ---------------------------------------------------------------------------
__global__ __launch_bounds__(256)
void pmwf_psd_accum(const float* __restrict__ xre, const float* __restrict__ xim,
                    const float* __restrict__ ms, const float* __restrict__ mn,
                    float* __restrict__ ws) {
    const int lane = threadIdx.x & 31;
    const int wave = blockIdx.x * (blockDim.x >> 5) + (threadIdx.x >> 5);
    if (wave >= NBF) return;                  // uniform per wave

    const int b    = wave >> 9;               // / FF
    const int f    = wave & (FF - 1);
    const int half = lane >> 4;               // 0: K=0,1   1: K=2,3
    const int r    = lane & 15;               // matrix row/col 0..15
    const int c    = r & 7;                   // channel

    const float* src = (r < 8 ? xre : xim) + ((size_t)(b * CC + c) * FF + f) * TT;
    const float* msp = ms + ((size_t)(b * FF + f)) * TT;
    const float* mnp = mn + ((size_t)(b * FF + f)) * TT;

    v8f acc_s = {0.f, 0.f, 0.f, 0.f, 0.f, 0.f, 0.f, 0.f};
    v8f acc_n = {0.f, 0.f, 0.f, 0.f, 0.f, 0.f, 0.f, 0.f};

    for (int t0 = 0; t0 < TT; t0 += 4) {
        const int tA = t0 + half * 2;
        v2f a   = *(const v2f*)(src + tA);     // val(r, tA), val(r, tA+1)
        v2f mvs = *(const v2f*)(msp + tA);
        v2f mvn = *(const v2f*)(mnp + tA);
        v2f bs = a * mvs;                      // B_s[k,n] = m_s[t]*val(n,t)
        v2f bn = a * mvn;                      // B_n[k,n] = m_n[t]*val(n,t)
        acc_s = __builtin_amdgcn_wmma_f32_16x16x4_f32(
            false, a, false, bs, (short)0, acc_s, false, false);
        acc_n = __builtin_amdgcn_wmma_f32_16x16x4_f32(
            false, a, false, bn, (short)0, acc_n, false, false);
    }

    // raw layout: ws[((wave*2 + mask)*32 + lane)*8 + vgpr]
    *(v8f*)(ws + ((size_t)(wave * 2 + 0) * 32 + lane) * 8) = acc_s;
    *(v8f*)(ws + ((size_t)(wave * 2 + 1) * 32 + lane) * 8) = acc_n;
}

// ---------------------------------------------------------------------------
// Stage 2: per-(b,f) finalize. Mask-sum normalization, diag regularization,
// complex Gauss-Jordan solve psd_n_reg^{-1} psd_s, trace normalization,
// store G[m][c] = conj(W[c][m]) interleaved (re,im).
// ---------------------------------------------------------------------------
__global__ __launch_bounds__(64)
void pmwf_solve(const float* __restrict__ ms, const float* __restrict__ mn,
                float* __restrict__ ws) {
    const int bf = blockIdx.x * blockDim.x + threadIdx.x;
    if (bf >= NBF) return;

    // mask sums (normalize=True with PSD_EPS)
    const float* msp = ms + (size_t)bf * TT;
    const float* mnp = mn + (size_t)bf * TT;
    float ssum = 0.f, nsum = 0.f;
    for (int t = 0; t < TT; ++t) { ssum += msp[t]; nsum += mnp[t]; }
    const float invS = 1.0f / (ssum + PSD_EPS_);
    const float invN = 1.0f / (nsum + PSD_EPS_);

    const float* rawS = ws + (size_t)(bf * 2 + 0) * 256;
    const float* rawN = ws + (size_t)(bf * 2 + 1) * 256;

    float Sre[8][8], Sim[8][8], Nre[8][8], Nim[8][8];
    // D element (M,N): vgpr = M%8, lane = (M>=8)*16 + N  ->  raw[lane*8 + vgpr]
    for (int cc = 0; cc < 8; ++cc)
        for (int e = 0; e < 8; ++e) {
            float d00 = rawS[e * 8 + cc];             // D[c,   e  ]
            float d11 = rawS[(24 + e) * 8 + cc];      // D[c+8, e+8]
            float d10 = rawS[(16 + e) * 8 + cc];      // D[c+8, e  ]
            float d01 = rawS[(8 + e) * 8 + cc];       // D[c,   e+8]
            Sre[cc][e] = (d00 + d11) * invS;
            Sim[cc][e] = (d10 - d01) * invS;
            d00 = rawN[e * 8 + cc];
            d11 = rawN[(24 + e) * 8 + cc];
            d10 = rawN[(16 + e) * 8 + cc];
            d01 = rawN[(8 + e) * 8 + cc];
            Nre[cc][e] = (d00 + d11) * invN;
            Nim[cc][e] = (d10 - d01) * invN;
        }

    // diagonal regularization: d = DIAG_REG*tr(psd_n).re + EPS
    float trn = 0.f;
    for (int i = 0; i < 8; ++i) trn += Nre[i][i];
    const float dreg = DIAG_REG_ * trn + EPS_;
    for (int i = 0; i < 8; ++i) Nre[i][i] += dreg;

    // complex Gauss-Jordan: S := N^{-1} S   (N is HPD after regularization)
    for (int k = 0; k < 8; ++k) {
        float pr = Nre[k][k], pi = Nim[k][k];
        float den = pr * pr + pi * pi;
        float ir = pr / den, ii = -pi / den;          // 1/pivot
        for (int j = 0; j < 8; ++j) {
            float nr = Nre[k][j], ni = Nim[k][j];
            Nre[k][j] = nr * ir - ni * ii;
            Nim[k][j] = nr * ii + ni * ir;
            float sr = Sre[k][j], si = Sim[k][j];
            Sre[k][j] = sr * ir - si * ii;
            Sim[k][j] = sr * ii + si * ir;
        }
        for (int i = 0; i < 8; ++i) {
            if (i == k) continue;
            float fr = Nre[i][k], fi = Nim[i][k];
            for (int j = 0; j < 8; ++j) {
                float nr = Nre[k][j], ni = Nim[k][j];
                Nre[i][j] -= fr * nr - fi * ni;
                Nim[i][j] -= fr * ni + fi * nr;
                float sr = Sre[k][j], si = Sim[k][j];
                Sre[i][j] -= fr * sr - fi * si;
                Sim[i][j] -= fr * si + fi * sr;
            }
        }
    }

    // W = X / (beta + tr(X))
    float trr = BETA_, tri = 0.f;
    for (int i = 0; i < 8; ++i) { trr += Sre[i][i]; tri += Sim[i][i]; }
    const float den = trr * trr + tri * tri;
    const float dr = trr / den, di = -tri / den;      // 1/(beta + tr)

    float* G = ws + G_OFF + (size_t)bf * 128;
    for (int m = 0; m < 8; ++m)
        for (int cc = 0; cc < 8; ++cc) {
            float wr = Sre[cc][m] * dr - Sim[cc][m] * di;
            float wi = Sre[cc][m] * di + Sim[cc][m] * dr;
            G[(m * 8 + cc) * 2 + 0] = wr;             // conj(W[c][m])
            G[(m * 8 + cc) * 2 + 1] = -wi;
        }
}

// ---------------------------------------------------------------------------
// Stage 3: out[b,m,f,t] = sum_c G[m][c] * x[c,t]. One block per (b,f);
// t contiguous across threads -> fully coalesced streaming (dominant bytes).
// ---------------------------------------------------------------------------
__global__ __launch_bounds__(256)
void pmwf_apply(const float* __restrict__ xre, const float* __restrict__ xim,
                const float* __restrict__ ws, float* __restrict__ out) {
    const int bf = blockIdx.x;
    const int b = bf >> 9, f = bf & (FF - 1);
    __shared__ float G[128];
    if (threadIdx.x < 128)
        G[threadIdx.x] = ws[G_OFF + (size_t)bf * 128 + threadIdx.x];
    __syncthreads();

    for (int t = threadIdx.x; t < TT; t += blockDim.x) {
        float xr[8], xi[8];
        for (int c = 0; c < 8; ++c) {
            size_t idx = ((size_t)(b * CC + c) * FF + f) * TT + t;
            xr[c] = xre[idx];
            xi[c] = xim[idx];
        }
        for (int m = 0; m < 8; ++m) {
            float orr = 0.f, oii = 0.f;
            for (int c = 0; c < 8; ++c) {
                float gr = G[(m * 8 + c) * 2 + 0];
                float gi = G[(m * 8 + c) * 2 + 1];
                orr += gr * xr[c] - gi * xi[c];
                oii += gr * xi[c] + gi * xr[c];
            }
            size_t oidx = ((size_t)(b * CC + m) * FF + f) * TT + t;
            float2 v; v.x = orr; v.y = oii;
            ((float2*)out)[oidx] = v;                 // complex64 interleaved
        }
    }
}

extern "C" void kernel_launch(void* const* d_in, const int* in_sizes, int n_in,
                              void* d_out, int out_size, void* d_ws, size_t ws_size,
                              hipStream_t stream) {
    (void)in_sizes; (void)n_in; (void)out_size; (void)ws_size;
    const float* xre = (const float*)d_in[0];
    const float* xim = (const float*)d_in[1];
    const float* ms  = (const float*)d_in[2];
    const float* mn  = (const float*)d_in[3];
    float* ws  = (float*)d_ws;
    float* out = (float*)d_out;

    // Stage 1: 2048 waves (1 per (b,f)), 8 waves per 256-thread block
    pmwf_psd_accum<<<NBF / 8, 256, 0, stream>>>(xre, xim, ms, mn, ws);
    // Stage 2: 1 thread per (b,f)
    pmwf_solve<<<NBF / 64, 64, 0, stream>>>(ms, mn, ws);
    // Stage 3: 1 block per (b,f)
    pmwf_apply<<<NBF, 256, 0, stream>>>(xre, xim, ws, out);
}